// GCNNet_19018115187322
// MI455X (gfx1250) — compile-verified
//
#include <hip/hip_runtime.h>
#include <math.h>
#include <stdint.h>

// ---------------------------------------------------------------------------
// 2-layer GCN for MI455X (gfx1250).
//  - Dense GEMMs use V_WMMA_F32_16X16X4_F32 (f32 in/out, K-step 4).
//  - GEMM1 X-tile staging uses GLOBAL_LOAD_ASYNC_TO_LDS_B128 when available
//    (ASYNCcnt-tracked DMA path), else falls back to ld/ds_store.
//  - Edge aggregation uses global_atomic_add_f32 (memory/L2-atomic bound).
// ---------------------------------------------------------------------------

typedef __attribute__((ext_vector_type(2))) float v2f;
typedef __attribute__((ext_vector_type(8))) float v8f;

#define NFEAT 500
#define HID   16
#define NLAB  40

#if defined(__has_builtin)
#if __has_builtin(__builtin_amdgcn_global_load_async_to_lds_b128) && \
    __has_builtin(__builtin_amdgcn_s_wait_asynccnt)
#define HAVE_ASYNC_LDS 1
#endif
#endif

#ifdef HAVE_ASYNC_LDS
// Builtin signature (from hipcc diagnostic):
//   void __builtin_amdgcn_global_load_async_to_lds_b128(
//       int4 AS(1)* gsrc, int4 AS(3)* ldst, imm offset, imm cpol)
typedef int async_v4i __attribute__((vector_size(16)));
typedef __attribute__((address_space(1))) async_v4i* gas_ptr;
typedef __attribute__((address_space(3))) async_v4i* las_ptr;
#endif

// ---------------------------------------------------------------------------
// Degree / normalization
// ---------------------------------------------------------------------------
__global__ void deg_init_kernel(float* __restrict__ deg, int N) {
    int i = blockIdx.x * blockDim.x + threadIdx.x;
    if (i < N) deg[i] = 1.0f;  // self-loop
}

__global__ void deg_edge_kernel(const int* __restrict__ dst, float* __restrict__ deg, int E) {
    int e = blockIdx.x * blockDim.x + threadIdx.x;
    if (e < E) atomicAdd(&deg[dst[e]], 1.0f);
}

__global__ void rsqrt_kernel(float* __restrict__ deg, int N) {
    int i = blockIdx.x * blockDim.x + threadIdx.x;
    if (i < N) deg[i] = rsqrtf(deg[i]);  // deg >= 1 always
}

// ---------------------------------------------------------------------------
// GEMM1: H1[N,16] = X[N,500] @ W1[500,16]   (f32 WMMA 16x16x4)
// 128 threads = 4 waves, 16 rows per wave, K staged in LDS chunks of 100.
// ---------------------------------------------------------------------------
__global__ void __launch_bounds__(128)
gemm1_kernel(const float* __restrict__ X, const float* __restrict__ W1,
             float* __restrict__ H1, int N) {
    __shared__ float Ws[NFEAT * HID];                     // 32000 B
    __shared__ __align__(16) float Xs[4][16][104];        // 100-col chunk, pad 4
    const int tid  = threadIdx.x;
    const int lane = tid & 31;
    const int wave = tid >> 5;
    const int row  = lane & 15;      // M (and N for B frag)
    const int khalf = lane >> 4;     // 0: K0,K1   1: K2,K3
    const int rowBase = blockIdx.x * 64 + wave * 16;

    for (int i = tid; i < NFEAT * HID; i += 128) Ws[i] = W1[i];
    __syncthreads();

    v8f acc = {};
    for (int c = 0; c < 5; ++c) {
        const int k0 = c * 100;
#ifdef HAVE_ASYNC_LDS
        // DMA 16 rows x 100 cols (25 float4 per row) straight into LDS.
        // All addresses 16B aligned: row stride 2000B, k0*4 = 400c B,
        // LDS row stride 416B.
        for (int idx = lane; idx < 16 * 25; idx += 32) {
            int r = idx / 25, q = idx - r * 25;
            int gr = rowBase + r;
            if (gr < N) {
                const float* gp = X + (size_t)gr * NFEAT + k0 + q * 4;
                float* lp = &Xs[wave][r][q * 4];
                __builtin_amdgcn_global_load_async_to_lds_b128(
                    (gas_ptr)(uintptr_t)gp, (las_ptr)(uintptr_t)lp, 0, 0);
            }
        }
        __builtin_amdgcn_s_wait_asynccnt(0);
        __syncthreads();
#else
        // coalesced stage of 16 rows x 100 cols of X into LDS
        for (int idx = lane; idx < 16 * 100; idx += 32) {
            int r = idx / 100, cc = idx - r * 100;
            int gr = rowBase + r;
            Xs[wave][r][cc] = (gr < N) ? X[(size_t)gr * NFEAT + k0 + cc] : 0.0f;
        }
        __syncthreads();
#endif
        for (int ks = 0; ks < 25; ++ks) {
            const int kk = ks * 4 + khalf * 2;           // chunk-local K
            v2f a, b;
            a.x = Xs[wave][row][kk];
            a.y = Xs[wave][row][kk + 1];
            b.x = Ws[(k0 + kk) * HID + row];             // B[K][N], N = lane&15
            b.y = Ws[(k0 + kk + 1) * HID + row];
            acc = __builtin_amdgcn_wmma_f32_16x16x4_f32(
                false, a, false, b, (short)0, acc, false, false);
        }
        __syncthreads();
    }
    // D layout: VGPR r -> M = r + 8*khalf, N = lane&15
    const int n = lane & 15;
    for (int r = 0; r < 8; ++r) {
        int m = rowBase + r + 8 * khalf;
        if (m < N) H1[(size_t)m * HID + n] = acc[r];
    }
}

// ---------------------------------------------------------------------------
// Aggregation layer 1 (HID feats): self-loop init + edge scatter-add
// ---------------------------------------------------------------------------
__global__ void self_init1_kernel(const float* __restrict__ H1, const float* __restrict__ dis,
                                  const float* __restrict__ b1, float* __restrict__ out, int N) {
    int t = blockIdx.x * blockDim.x + threadIdx.x;
    if (t < N * HID) {
        int i = t >> 4, f = t & 15;
        float d = dis[i];
        out[t] = b1[f] + d * d * H1[t];
    }
}

__global__ void scatter1_kernel(const int* __restrict__ src, const int* __restrict__ dst,
                                const float* __restrict__ H1, const float* __restrict__ dis,
                                float* __restrict__ out, int E) {
    int t = blockIdx.x * blockDim.x + threadIdx.x;
    if (t < E * HID) {
        int e = t >> 4, f = t & 15;
        int s = src[e], d = dst[e];
        float v = dis[s] * dis[d] * H1[(size_t)s * HID + f];
        atomicAdd(&out[(size_t)d * HID + f], v);
    }
}

// ---------------------------------------------------------------------------
// GEMM2: H2[N,40] = relu(AG1)[N,16] @ W2[16,40]   (fused ReLU on A loads)
// 256 threads = 8 waves, 16 rows/wave, 3 column tiles of 16 (N=40 padded).
// ---------------------------------------------------------------------------
__global__ void __launch_bounds__(256)
gemm2_kernel(const float* __restrict__ AG1, const float* __restrict__ W2,
             float* __restrict__ H2, int N) {
    __shared__ float Ws[16 * 48];            // W2 zero-padded to 48 cols
    const int tid  = threadIdx.x;
    const int lane = tid & 31;
    const int wave = tid >> 5;
    const int row  = lane & 15;
    const int khalf = lane >> 4;
    const int rowBase = blockIdx.x * 128 + wave * 16;

    for (int i = tid; i < 16 * 48; i += 256) Ws[i] = 0.0f;
    __syncthreads();
    for (int i = tid; i < 16 * NLAB; i += 256) {
        int k = i / NLAB, cc = i - k * NLAB;
        Ws[k * 48 + cc] = W2[i];
    }
    __syncthreads();

    v8f acc0 = {}, acc1 = {}, acc2 = {};
    const int gr = rowBase + row;
    for (int ks = 0; ks < 4; ++ks) {
        const int kk = ks * 4 + khalf * 2;
        v2f a, b0, b1, b2;
        float x0 = (gr < N) ? AG1[(size_t)gr * HID + kk] : 0.0f;
        float x1 = (gr < N) ? AG1[(size_t)gr * HID + kk + 1] : 0.0f;
        a.x = fmaxf(x0, 0.0f);           // fused ReLU
        a.y = fmaxf(x1, 0.0f);
        b0.x = Ws[kk * 48 + row];        b0.y = Ws[(kk + 1) * 48 + row];
        b1.x = Ws[kk * 48 + 16 + row];   b1.y = Ws[(kk + 1) * 48 + 16 + row];
        b2.x = Ws[kk * 48 + 32 + row];   b2.y = Ws[(kk + 1) * 48 + 32 + row];
        acc0 = __builtin_amdgcn_wmma_f32_16x16x4_f32(false, a, false, b0, (short)0, acc0, false, false);
        acc1 = __builtin_amdgcn_wmma_f32_16x16x4_f32(false, a, false, b1, (short)0, acc1, false, false);
        acc2 = __builtin_amdgcn_wmma_f32_16x16x4_f32(false, a, false, b2, (short)0, acc2, false, false);
    }
    const int n = lane & 15;
    for (int r = 0; r < 8; ++r) {
        int m = rowBase + r + 8 * khalf;
        if (m < N) {
            float* o = &H2[(size_t)m * NLAB];
            o[n]      = acc0[r];
            o[16 + n] = acc1[r];
            if (32 + n < NLAB) o[32 + n] = acc2[r];
        }
    }
}

// ---------------------------------------------------------------------------
// Aggregation layer 2 (NLAB feats)
// ---------------------------------------------------------------------------
__global__ void self_init2_kernel(const float* __restrict__ H2, const float* __restrict__ dis,
                                  const float* __restrict__ b2, float* __restrict__ out, int N) {
    int t = blockIdx.x * blockDim.x + threadIdx.x;
    if (t < N * NLAB) {
        int i = t / NLAB, f = t - i * NLAB;
        float d = dis[i];
        out[t] = b2[f] + d * d * H2[t];
    }
}

__global__ void scatter2_kernel(const int* __restrict__ src, const int* __restrict__ dst,
                                const float* __restrict__ H2, const float* __restrict__ dis,
                                float* __restrict__ out, int E) {
    long long t = (long long)blockIdx.x * blockDim.x + threadIdx.x;
    if (t < (long long)E * NLAB) {
        int e = (int)(t / NLAB), f = (int)(t - (long long)e * NLAB);
        int s = src[e], d = dst[e];
        float v = dis[s] * dis[d] * H2[(size_t)s * NLAB + f];
        atomicAdd(&out[(size_t)d * NLAB + f], v);
    }
}

// ---------------------------------------------------------------------------
// log_softmax over 40 labels: one wave32 per node, shfl_xor reductions
// ---------------------------------------------------------------------------
__global__ void __launch_bounds__(256)
lsm_kernel(const float* __restrict__ A, float* __restrict__ out, int N) {
    const int lane = threadIdx.x & 31;
    const int wave = threadIdx.x >> 5;
    const int node = blockIdx.x * 8 + wave;
    if (node >= N) return;
    const float* r = &A[(size_t)node * NLAB];
    float x0 = r[lane];                                   // lanes 0..31
    float x1 = (lane < NLAB - 32) ? r[32 + lane] : -INFINITY;
    float m = fmaxf(x0, x1);
    for (int o = 16; o > 0; o >>= 1) m = fmaxf(m, __shfl_xor(m, o, 32));
    float s = expf(x0 - m) + ((lane < NLAB - 32) ? expf(x1 - m) : 0.0f);
    for (int o = 16; o > 0; o >>= 1) s += __shfl_xor(s, o, 32);
    float l = m + logf(s);
    out[(size_t)node * NLAB + lane] = x0 - l;
    if (lane < NLAB - 32) out[(size_t)node * NLAB + 32 + lane] = x1 - l;
}

// ---------------------------------------------------------------------------
// launch
// ---------------------------------------------------------------------------
extern "C" void kernel_launch(void* const* d_in, const int* in_sizes, int n_in,
                              void* d_out, int out_size, void* d_ws, size_t ws_size,
                              hipStream_t stream) {
    const float* x  = (const float*)d_in[0];
    const int*   ei = (const int*)d_in[1];      // [2, E] int
    const float* W1 = (const float*)d_in[2];
    const float* b1 = (const float*)d_in[3];
    const float* W2 = (const float*)d_in[4];
    const float* b2 = (const float*)d_in[5];

    const int N = in_sizes[0] / NFEAT;          // 100000
    const int E = in_sizes[1] / 2;              // 3200000
    const int* src = ei;
    const int* dst = ei + E;

    // workspace layout (floats)
    float* ws   = (float*)d_ws;
    float* dis  = ws;                            // N        (deg -> dis in place)
    float* h1   = dis  + 131072;                 // N*16
    float* ag1  = h1   + (size_t)N * HID;        // N*16
    float* h2   = ag1  + (size_t)N * HID;        // N*40
    float* ag2  = h2   + (size_t)N * NLAB;       // N*40

    // --- degree / normalization ---
    deg_init_kernel<<<(N + 255) / 256, 256, 0, stream>>>(dis, N);
    deg_edge_kernel<<<(E + 255) / 256, 256, 0, stream>>>(dst, dis, E);
    rsqrt_kernel<<<(N + 255) / 256, 256, 0, stream>>>(dis, N);

    // --- layer 1 ---
    gemm1_kernel<<<(N + 63) / 64, 128, 0, stream>>>(x, W1, h1, N);
    self_init1_kernel<<<(N * HID + 255) / 256, 256, 0, stream>>>(h1, dis, b1, ag1, N);
    scatter1_kernel<<<(E * HID + 255) / 256, 256, 0, stream>>>(src, dst, h1, dis, ag1, E);

    // --- layer 2 (ReLU fused into GEMM2 A loads) ---
    gemm2_kernel<<<(N + 127) / 128, 256, 0, stream>>>(ag1, W2, h2, N);
    self_init2_kernel<<<(N * NLAB + 255) / 256, 256, 0, stream>>>(h2, dis, b2, ag2, N);
    {
        long long tot = (long long)E * NLAB;
        int blocks = (int)((tot + 255) / 256);
        scatter2_kernel<<<blocks, 256, 0, stream>>>(src, dst, h2, dis, ag2, E);
    }

    // --- log_softmax ---
    lsm_kernel<<<(N + 7) / 8, 256, 0, stream>>>(ag2, (float*)d_out, N);
}